// Attention_37056977829929
// MI455X (gfx1250) — compile-verified
//
#include <hip/hip_runtime.h>

// Problem constants (match reference)
#define PRO_NUM  64
#define MAX_LEN  1024
#define HID      256
#define TERM_NUM 512

typedef __attribute__((ext_vector_type(2))) float v2f;
typedef __attribute__((ext_vector_type(4))) float v4f;
typedef __attribute__((ext_vector_type(8))) float v8f;

// LDS row strides padded by 4 floats -> conflict-free b64 A-fragment reads
#define SQ_STRIDE 260    // 256 + 4
#define SS_STRIDE 1028   // 1024 + 4

// D = A(16x4 f32) x B(4x16 f32) + C(16x16 f32), full fp32 matrix pipe.
// Probe-documented signature pattern: (neg_a, A, neg_b, B, c_mod, C, reuse_a, reuse_b)
static __device__ __forceinline__ v8f wmma_f32(v2f a, v2f b, v8f c) {
    return __builtin_amdgcn_wmma_f32_16x16x4_f32(false, a, false, b, (short)0, c,
                                                 false, false);
}

__global__ __launch_bounds__(256)
void attn_fp32_wmma_kernel(const float* __restrict__ term,  // [512,256]
                           const float* __restrict__ pro,   // [64,1024,256]
                           const int*   __restrict__ lens,  // [64]
                           float* __restrict__ out,         // [64,512,256]
                           float* __restrict__ attn)        // [64,512,1024]
{
    extern __shared__ float smem[];
    float* sQ = smem;                        // [16][SQ_STRIDE]  Q tile
    float* sS = smem + 16 * SQ_STRIDE;       // [16][SS_STRIDE]  scores -> attn tile

    const int p    = blockIdx.y;
    const int t0   = blockIdx.x * 16;        // 16 terms per workgroup
    const int tid  = threadIdx.x;
    const int lane = tid & 31;
    const int wave = tid >> 5;               // 8 waves
    const int hi   = lane >> 4;              // half-wave select (0/1)
    const int m    = lane & 15;              // A-row / B-col index

    const float* proP = pro + (size_t)p * MAX_LEN * HID;

    // ---------------- Stage Q tile (16 x 256) into LDS, float4 coalesced ----------
    #pragma unroll
    for (int j = 0; j < 4; ++j) {
        int idx = tid + 256 * j;             // 1024 float4's total
        int row = idx >> 6;                  // 64 float4 per row
        int col = (idx & 63) << 2;
        v4f q = *(const v4f*)(term + (size_t)(t0 + row) * HID + col);
        *(v4f*)(sQ + row * SQ_STRIDE + col) = q;
    }
    __syncthreads();

    // ---------------- Phase 1: scores(16 x 1024) = Q * K^T -----------------------
    // Wave w covers L columns [w*128, w*128+128) as 8 WMMA tiles; K-loop over H.
    {
        const int l0w = wave * 128;
        v8f acc[8] = {};
        for (int kb = 0; kb < HID; kb += 4) {
            // A fragment: A[m][kb + 2*hi .. +1], reused across all 8 N-tiles
            v2f a = *(const v2f*)(sQ + m * SQ_STRIDE + kb + 2 * hi);
            #pragma unroll
            for (int tI = 0; tI < 8; ++tI) {
                // B fragment: B[k][n] = pro[p, l0+n, kb+k] -> contiguous h pair
                v2f b = *(const v2f*)(proP + (size_t)(l0w + tI * 16 + m) * HID
                                            + kb + 2 * hi);
                acc[tI] = wmma_f32(a, b, acc[tI]);
            }
        }
        // C/D layout: row = r + 8*hi, col = m
        #pragma unroll
        for (int tI = 0; tI < 8; ++tI) {
            #pragma unroll
            for (int r = 0; r < 8; ++r) {
                sS[(r + 8 * hi) * SS_STRIDE + l0w + tI * 16 + m] = acc[tI][r];
            }
        }
    }
    __syncthreads();

    // ---------------- Masked softmax over L (2 rows per wave) --------------------
    {
        const int len = lens[p];             // in [1, MAX_LEN] -> row never all-masked
        #pragma unroll
        for (int rr = 0; rr < 2; ++rr) {
            const int r = wave * 2 + rr;
            float vals[32];
            float mx = -__builtin_inff();
            #pragma unroll
            for (int i = 0; i < 32; ++i) {
                int l = lane + 32 * i;
                float s = (l < len) ? sS[r * SS_STRIDE + l] : -__builtin_inff();
                vals[i] = s;
                mx = fmaxf(mx, s);
            }
            #pragma unroll
            for (int off = 16; off > 0; off >>= 1)
                mx = fmaxf(mx, __shfl_xor(mx, off, 32));
            float sum = 0.f;
            #pragma unroll
            for (int i = 0; i < 32; ++i) {
                float e = __expf(vals[i] - mx);   // masked: exp(-inf) = 0
                vals[i] = e;
                sum += e;
            }
            #pragma unroll
            for (int off = 16; off > 0; off >>= 1)
                sum += __shfl_xor(sum, off, 32);
            float inv = 1.0f / sum;
            float* attnRow = attn + ((size_t)p * TERM_NUM + t0 + r) * MAX_LEN;
            #pragma unroll
            for (int i = 0; i < 32; ++i) {
                int l = lane + 32 * i;
                float aV = vals[i] * inv;
                sS[r * SS_STRIDE + l] = aV;      // reuse LDS tile as attn A-matrix
                attnRow[l] = aV;                 // mandatory attn output (coalesced)
            }
        }
    }
    __syncthreads();

    // ---------------- Phase 2: out(16 x 256) = attn * V --------------------------
    // Wave w covers H columns [w*32, w*32+32) as 2 WMMA tiles; K-loop over L=1024.
    {
        const int h0 = wave * 32;
        v8f acc[2] = {};
        for (int kb = 0; kb < MAX_LEN; kb += 4) {
            v2f a = *(const v2f*)(sS + m * SS_STRIDE + kb + 2 * hi);
            const int l = kb + 2 * hi;
            #pragma unroll
            for (int tI = 0; tI < 2; ++tI) {
                // B[k][n] = pro[p, l, h0+n] : 64B-coalesced per half-wave
                v2f b;
                b.x = proP[(size_t)l * HID + h0 + tI * 16 + m];
                b.y = proP[(size_t)(l + 1) * HID + h0 + tI * 16 + m];
                acc[tI] = wmma_f32(a, b, acc[tI]);
            }
        }
        #pragma unroll
        for (int tI = 0; tI < 2; ++tI) {
            #pragma unroll
            for (int r = 0; r < 8; ++r) {
                out[((size_t)p * TERM_NUM + t0 + r + 8 * hi) * HID
                    + h0 + tI * 16 + m] = acc[tI][r];
            }
        }
    }
}

extern "C" void kernel_launch(void* const* d_in, const int* in_sizes, int n_in,
                              void* d_out, int out_size, void* d_ws, size_t ws_size,
                              hipStream_t stream) {
    (void)in_sizes; (void)n_in; (void)out_size; (void)d_ws; (void)ws_size;
    const float* term = (const float*)d_in[0];
    const float* pro  = (const float*)d_in[1];
    const int*   lens = (const int*)d_in[2];

    float* out  = (float*)d_out;                                     // [64,512,256]
    float* attn = out + (size_t)PRO_NUM * TERM_NUM * HID;            // [64,512,1024]

    dim3 grid(TERM_NUM / 16, PRO_NUM);   // 32 term-tiles x 64 proteins
    dim3 block(256);                     // 8 wave32 waves
    size_t shmem = (size_t)(16 * SQ_STRIDE + 16 * SS_STRIDE) * sizeof(float); // 82432 B

    attn_fp32_wmma_kernel<<<grid, block, shmem, stream>>>(term, pro, lens, out, attn);
}